// retmil_11768210391318
// MI455X (gfx1250) — compile-verified
//
#include <hip/hip_runtime.h>
#include <hip/hip_bf16.h>
#include <math.h>

typedef __attribute__((ext_vector_type(2))) float v2f;
typedef __attribute__((ext_vector_type(8))) float v8f;
typedef __attribute__((ext_vector_type(4))) unsigned int v4u;
typedef __attribute__((ext_vector_type(4))) int v4i;
typedef __attribute__((ext_vector_type(8))) int v8i;

#define D_MODEL   256
#define N_HEADS   8
#define HEAD_DIM  32

// -------- CDNA5 WMMA: D = A(16x4 f32) x B(4x16 f32) + C(16x16 f32) --------
// A frag: 2 floats/lane (lane m = l&15; k = 2*(l>>4) + {0,1})
// B frag: 2 floats/lane (lane n = l&15; k = 2*(l>>4) + {0,1})
// C/D   : 8 floats/lane (m = r + 8*(l>>4), n = l&15)
__device__ __forceinline__ v8f wmma_f32x4(v2f a, v2f b, v8f c) {
  return __builtin_amdgcn_wmma_f32_16x16x4_f32(false, a, false, b, (short)0, c,
                                               false, false);
}

// ---------------------------------------------------------------------------
// Tensor Data Mover: DMA one contiguous 16 KB block (4096 f32) into LDS.
// D# built per CDNA5 ISA ch.8: group0 = {count, lds_addr, global_addr, type=2},
// group1 = {data_size=4B, tensor_dim0=4096, dim1=1, tile_dim0=4096,
//           tile_dim1=1, dim0_stride=4096}.  Issued by one wave; completion
// tracked with TENSORcnt.
// ---------------------------------------------------------------------------
#if defined(__has_builtin)
#if __has_builtin(__builtin_amdgcn_tensor_load_to_lds)
#define HAVE_TDM 1
#endif
#endif

__device__ __forceinline__ void tdm_load_16kb(const void* gsrc,
                                              unsigned lds_off) {
#ifdef HAVE_TDM
  const unsigned long long ga = (unsigned long long)(uintptr_t)gsrc;
  v4u g0 = {1u,  // count=1, user descriptor, no gather
            lds_off,
            (unsigned)(ga & 0xffffffffu),
            (unsigned)((ga >> 32) & 0x01ffffffu) | (2u << 30)};  // type=2
  v8i g1 = {(int)(2u << 16),     // [17:16] data_size = 2 (4 bytes)
            (int)(4096u << 16),  // [79:48] tensor_dim0 = 4096 (lo16 here)
            (int)(1u << 16),     // dim0 hi16 = 0 | [111:80] tensor_dim1 = 1
            (int)(4096u << 16),  // dim1 hi16 = 0 | [127:112] tile_dim0 = 4096
            1,                   // [143:128] tile_dim1 = 1, tile_dim2 = 0
            4096,                // [207:160] tensor_dim0_stride = 4096
            0, 0};
  v4i gz = {0, 0, 0, 0};
#if defined(__clang_major__) && (__clang_major__ >= 23)
  v8i gz8 = {0, 0, 0, 0, 0, 0, 0, 0};
  __builtin_amdgcn_tensor_load_to_lds(g0, g1, gz, gz, gz8, 0);
#else
  __builtin_amdgcn_tensor_load_to_lds(g0, g1, gz, gz, 0);
#endif
  __builtin_amdgcn_s_wait_tensorcnt(0);
#else
  (void)gsrc; (void)lds_off;
#endif
}

// ---------------------------------------------------------------------------
// GEMM: Y[M,256] = X[M,256] @ W^T + bias.  W is [256,256] row-major (reference
// stores weights as (out,in), y = x @ W.T).  grid = (M/16, 2), block = 256.
// HEAD_MAJOR==1 scatters into [bseq, head, n, hd] (bseq = row / N_SEQ, a shift
// since N_SEQ is a compile-time power of two).
// ---------------------------------------------------------------------------
template <int N_SEQ, int HEAD_MAJOR>
__global__ __launch_bounds__(256) void gemm256_bias(
    const float* __restrict__ X, const float* __restrict__ W,
    const float* __restrict__ bias, float* __restrict__ Y) {
  __shared__ float As[16 * 256];
  const int tid = threadIdx.x;
  const size_t row0 = (size_t)blockIdx.x * 16;

  // stage the 16x256 A tile (contiguous 16 KB) into LDS
#ifdef HAVE_TDM
  if (tid < 32) {  // wave 0 issues the tensor DMA and waits on TENSORcnt
    tdm_load_16kb(X + row0 * 256, (unsigned)(uintptr_t)As);
  }
#else
  {
    const float* src = X + row0 * 256;
#pragma unroll
    for (int i = 0; i < 16; ++i) As[tid + 256 * i] = src[tid + 256 * i];
  }
#endif
  __syncthreads();

  const int lane   = tid & 31;
  const int wave   = tid >> 5;
  const int tile_o = blockIdx.y * 128 + wave * 16;
  const int mrow   = lane & 15;
  const int khalf  = (lane >> 4) * 2;

  const float* wrow = W + (size_t)(tile_o + mrow) * 256;  // B lane n == mrow
  v8f acc = {};
#pragma unroll 8
  for (int kc = 0; kc < 64; ++kc) {
    const int koff = kc * 4 + khalf;
    v2f a = *(const v2f*)(As + mrow * 256 + koff);
    v2f b = *(const v2f*)(wrow + koff);
    acc = wmma_f32x4(a, b, acc);
  }

  const int ncol = tile_o + mrow;
  const float bv = bias[ncol];
#pragma unroll
  for (int r = 0; r < 8; ++r) {
    const int m = r + (lane >> 4) * 8;
    const size_t row = row0 + m;
    const float val = acc[r] + bv;
    if (HEAD_MAJOR) {
      const size_t bs = row / (size_t)N_SEQ;  // compile-time shift
      const size_t nn = row & (size_t)(N_SEQ - 1);
      const int h = ncol >> 5;
      const int c = ncol & 31;
      Y[((bs * N_HEADS + h) * (size_t)N_SEQ + nn) * HEAD_DIM + c] = val;
    } else {
      Y[row * D_MODEL + ncol] = val;
    }
  }
}

// ---------------------------------------------------------------------------
// Retention attention for one 16-row i-tile of one (seq, head):
//   s = (Q K^T) * M,  M[i,j] = exp(-(j-i)) for j>=i else 0;  o = s V.
// Q,K,V are head-major [(bseq*8), N_SEQ, 32].  O is token-major [bseq,n,256].
// grid = bseq*8*(N_SEQ/16) blocks of 1 wave (32 threads).
// ---------------------------------------------------------------------------
template <int N_SEQ>
__global__ __launch_bounds__(32) void retention_attn(
    const float* __restrict__ Q, const float* __restrict__ K,
    const float* __restrict__ V, float* __restrict__ O) {
  __shared__ float sT[256];  // masked score tile, C-layout -> A-layout bridge

  constexpr int ITILES = N_SEQ >> 4;
  const int it = blockIdx.x % ITILES;        // compile-time modulus
  const int sh = blockIdx.x / ITILES;        // bs*8 + h
  const int h  = sh & 7;
  const int bs = sh >> 3;

  const int lane  = threadIdx.x;
  const int mrow  = lane & 15;
  const int khalf = (lane >> 4) * 2;
  const int ig    = it * 16;

  const float* qp = Q + (size_t)sh * N_SEQ * HEAD_DIM;
  const float* kp = K + (size_t)sh * N_SEQ * HEAD_DIM;
  const float* vp = V + (size_t)sh * N_SEQ * HEAD_DIM;

  // Q fragments for the whole hd=32 K-dimension stay in registers
  v2f qa[8];
#pragma unroll
  for (int kc = 0; kc < 8; ++kc)
    qa[kc] = *(const v2f*)(qp + (size_t)(ig + mrow) * HEAD_DIM + kc * 4 + khalf);

  v8f o0 = {}, o1 = {};

  // mask is zero for whole tiles below the diagonal -> start at jt = it
  for (int jt = it; jt < ITILES; ++jt) {
    const int jg = jt * 16;

    // s_tile = Q_tile (16x32) x K_tile^T (32x16)
    v8f sc = {};
#pragma unroll
    for (int kc = 0; kc < 8; ++kc) {
      v2f bk = *(const v2f*)(kp + (size_t)(jg + mrow) * HEAD_DIM + kc * 4 + khalf);
      sc = wmma_f32x4(qa[kc], bk, sc);
    }

    // apply decay mask elementwise (C layout) and stash to LDS
#pragma unroll
    for (int r = 0; r < 8; ++r) {
      const int m  = r + (lane >> 4) * 8;
      const int dd = (jg + mrow) - (ig + m);  // j - i
      sT[m * 16 + mrow] = (dd >= 0) ? sc[r] * __expf(-(float)dd) : 0.0f;
    }
    __syncthreads();  // cross-lane C-layout -> A-layout handoff

    // o_tile += s_tile (16x16) x V_tile (16x32), split into two 16-col halves
#pragma unroll
    for (int kc = 0; kc < 4; ++kc) {
      const int kof = kc * 4 + khalf;
      v2f af = *(const v2f*)(sT + mrow * 16 + kof);
      const float* vr = vp + (size_t)(jg + kof) * HEAD_DIM;
      v2f b0, b1;
      b0[0] = vr[mrow];               b0[1] = vr[HEAD_DIM + mrow];
      b1[0] = vr[16 + mrow];          b1[1] = vr[HEAD_DIM + 16 + mrow];
      o0 = wmma_f32x4(af, b0, o0);
      o1 = wmma_f32x4(af, b1, o1);
    }
    __syncthreads();  // protect sT before next iteration overwrites it
  }

#pragma unroll
  for (int r = 0; r < 8; ++r) {
    const int m = r + (lane >> 4) * 8;
    const size_t base =
        ((size_t)bs * N_SEQ + (ig + m)) * D_MODEL + h * HEAD_DIM;
    O[base + mrow]      = o0[r];
    O[base + 16 + mrow] = o1[r];
  }
}

// ---------------------------------------------------------------------------
// GroupNorm(groups=1): per batch element, normalize over (n_seq, 256).
// grid = nseq, block = 256.
// ---------------------------------------------------------------------------
__global__ __launch_bounds__(256) void groupnorm_k(
    const float* __restrict__ X, float* __restrict__ Y,
    const float* __restrict__ w, const float* __restrict__ b, int n_seq) {
  __shared__ float rs_[256];
  __shared__ float rq_[256];
  const int tid = threadIdx.x;
  const int total = n_seq * D_MODEL;
  const size_t base = (size_t)blockIdx.x * total;

  float s = 0.f, ss = 0.f;
  for (int i = tid; i < total; i += 256) {
    const float v = X[base + i];
    s += v;
    ss += v * v;
  }
  rs_[tid] = s;
  rq_[tid] = ss;
  __syncthreads();
  for (int st = 128; st > 0; st >>= 1) {
    if (tid < st) {
      rs_[tid] += rs_[tid + st];
      rq_[tid] += rq_[tid + st];
    }
    __syncthreads();
  }
  const float inv_n = 1.0f / (float)total;
  const float mu  = rs_[0] * inv_n;
  const float var = rq_[0] * inv_n - mu * mu;  // biased, matches jnp.var
  const float rsg = rsqrtf(var + 1e-5f);
  const float wd = w[tid];  // feature index == tid (stride-256 loop)
  const float bd = b[tid];
  for (int i = tid; i < total; i += 256)
    Y[base + i] = (X[base + i] - mu) * rsg * wd + bd;
}

// a_logits = g * tanh(u1) * sigmoid(u2), written in-place over u1
__global__ __launch_bounds__(256) void gate_logits_k(
    float* __restrict__ u1, const float* __restrict__ u2,
    const float* __restrict__ g, size_t total) {
  const size_t i = (size_t)blockIdx.x * 256 + threadIdx.x;
  if (i < total) {
    const float sg = 1.0f / (1.0f + __expf(-u2[i]));
    u1[i] = g[0] * tanhf(u1[i]) * sg;
  }
}

// column softmax over n_seq (axis=1) then weighted sum with x.
// grid = nseq, block = 256 (one thread per feature).
__global__ __launch_bounds__(256) void softmax_pool_k(
    const float* __restrict__ T, const float* __restrict__ X,
    float* __restrict__ OUT, int n_seq) {
  const int d = threadIdx.x;
  const size_t base = (size_t)blockIdx.x * n_seq * D_MODEL + d;
  float m = -3.4e38f;
  for (int n = 0; n < n_seq; ++n)
    m = fmaxf(m, T[base + (size_t)n * D_MODEL]);
  float sum = 0.f, ws = 0.f;
  for (int n = 0; n < n_seq; ++n) {
    const size_t idx = base + (size_t)n * D_MODEL;
    const float e = __expf(T[idx] - m);
    sum += e;
    ws += e * X[idx];
  }
  OUT[(size_t)blockIdx.x * D_MODEL + d] = ws / sum;
}

// BatchNorm over batch-of-4 (training stats, biased var) + 256->2 classifier
__global__ __launch_bounds__(256) void bn_cls_k(
    const float* __restrict__ emb, const float* __restrict__ bnw,
    const float* __restrict__ bnb, const float* __restrict__ clsW,
    const float* __restrict__ clsb, float* __restrict__ out) {
  __shared__ float z[4 * 256];
  const int d = threadIdx.x;
  const float e0 = emb[d], e1 = emb[256 + d], e2 = emb[512 + d],
              e3 = emb[768 + d];
  const float mu = 0.25f * (e0 + e1 + e2 + e3);
  const float d0 = e0 - mu, d1 = e1 - mu, d2 = e2 - mu, d3 = e3 - mu;
  const float var = 0.25f * (d0 * d0 + d1 * d1 + d2 * d2 + d3 * d3);
  const float rs = rsqrtf(var + 1e-5f);
  const float w = bnw[d], b = bnb[d];
  z[d]       = d0 * rs * w + b;
  z[256 + d] = d1 * rs * w + b;
  z[512 + d] = d2 * rs * w + b;
  z[768 + d] = d3 * rs * w + b;
  __syncthreads();
  if (d < 8) {
    const int bb = d >> 1, j = d & 1;
    float acc = clsb[j];
    for (int k = 0; k < 256; ++k) acc += z[bb * 256 + k] * clsW[j * 256 + k];
    out[bb * 2 + j] = acc;
  }
}

// ---------------------------------------------------------------------------
extern "C" void kernel_launch(void* const* d_in, const int* in_sizes, int n_in,
                              void* d_out, int out_size, void* d_ws,
                              size_t ws_size, hipStream_t stream) {
  (void)in_sizes; (void)n_in; (void)out_size; (void)ws_size;
  auto F = [&](int i) { return (const float*)d_in[i]; };
  const float* bags = F(0);
  // input order: bags, r1[WQ,bQ,WK,bK,WV,bV,gnw,gnb], r2[...], rg[...],
  //              pl[W,bW,U,bU,g], pg[...], bn_w, bn_b, cls_W, cls_b

  const size_t BIG = (size_t)128 * 512 * 256;  // 16.78M floats
  float* ws = (float*)d_ws;
  float* q  = ws;
  float* k  = q + BIG;
  float* v  = k + BIG;
  float* o  = v + BIG;
  float* x1 = o + BIG;
  float* x2 = x1 + BIG;
  float* local = x2 + BIG;            // 128*256
  float* qg  = local + 32768;         // 4*8*32*32 = 32768
  float* kg  = qg + 32768;
  float* vg  = kg + 32768;
  float* og  = vg + 32768;
  float* gg  = og + 32768;
  float* emb = gg + 32768;            // 4*256

  const dim3 gBig(4096, 2);   // M = 65536 rows
  const dim3 gSml(8, 2);      // M = 128 rows
  const dim3 blk(256);

  // ---- retention layer 1 (x = bags viewed as [128*512, 256]) ----
  gemm256_bias<512, 1><<<gBig, blk, 0, stream>>>(bags, F(1), F(2), q);
  gemm256_bias<512, 1><<<gBig, blk, 0, stream>>>(bags, F(3), F(4), k);
  gemm256_bias<512, 1><<<gBig, blk, 0, stream>>>(bags, F(5), F(6), v);
  retention_attn<512><<<128 * 8 * 32, 32, 0, stream>>>(q, k, v, o);
  groupnorm_k<<<128, blk, 0, stream>>>(o, x1, F(7), F(8), 512);

  // ---- retention layer 2 ----
  gemm256_bias<512, 1><<<gBig, blk, 0, stream>>>(x1, F(9),  F(10), q);
  gemm256_bias<512, 1><<<gBig, blk, 0, stream>>>(x1, F(11), F(12), k);
  gemm256_bias<512, 1><<<gBig, blk, 0, stream>>>(x1, F(13), F(14), v);
  retention_attn<512><<<128 * 8 * 32, 32, 0, stream>>>(q, k, v, o);
  groupnorm_k<<<128, blk, 0, stream>>>(o, x2, F(15), F(16), 512);

  // ---- local attention pool: (128,512,256) -> (128,256) ----
  gemm256_bias<512, 0><<<gBig, blk, 0, stream>>>(x2, F(25), F(26), q);  // u1
  gemm256_bias<512, 0><<<gBig, blk, 0, stream>>>(x2, F(27), F(28), k);  // u2
  gate_logits_k<<<(unsigned)(BIG / 256), blk, 0, stream>>>(q, k, F(29), BIG);
  softmax_pool_k<<<128, blk, 0, stream>>>(q, x2, local, 512);

  // ---- global retention on (4,32,256) ----
  gemm256_bias<32, 1><<<gSml, blk, 0, stream>>>(local, F(17), F(18), qg);
  gemm256_bias<32, 1><<<gSml, blk, 0, stream>>>(local, F(19), F(20), kg);
  gemm256_bias<32, 1><<<gSml, blk, 0, stream>>>(local, F(21), F(22), vg);
  retention_attn<32><<<4 * 8 * 2, 32, 0, stream>>>(qg, kg, vg, og);
  groupnorm_k<<<4, blk, 0, stream>>>(og, gg, F(23), F(24), 32);

  // ---- global attention pool: (4,32,256) -> (4,256) ----
  gemm256_bias<32, 0><<<gSml, blk, 0, stream>>>(gg, F(30), F(31), qg);
  gemm256_bias<32, 0><<<gSml, blk, 0, stream>>>(gg, F(32), F(33), kg);
  gate_logits_k<<<128, blk, 0, stream>>>(qg, kg, F(34), (size_t)32768);
  softmax_pool_k<<<4, blk, 0, stream>>>(qg, gg, emb, 32);

  // ---- batchnorm (batch stats) + classifier -> d_out[4,2] ----
  bn_cls_k<<<1, blk, 0, stream>>>(emb, F(35), F(36), F(37), F(38),
                                  (float*)d_out);
}